// ChebNet_2903397892894
// MI455X (gfx1250) — compile-verified
//
#include <hip/hip_runtime.h>
#include <hip/hip_bf16.h>
#include <math.h>

typedef __attribute__((ext_vector_type(2))) float v2f;
typedef __attribute__((ext_vector_type(8))) float v8f;

#define N_NODES 100000
#define N_EDGES 1600000

// ---------------- elementwise / degree kernels ----------------

__global__ void deg_kernel(const long long* __restrict__ dst,
                           float* __restrict__ deg, int e) {
    int i = blockIdx.x * blockDim.x + threadIdx.x;
    if (i < e) atomicAdd(&deg[(int)dst[i]], 1.0f);
}

__global__ void dis_kernel(const float* __restrict__ deg,
                           float* __restrict__ dis, int n) {
    int i = blockIdx.x * blockDim.x + threadIdx.x;
    if (i < n) {
        float d = deg[i];
        dis[i] = (d > 0.0f) ? rsqrtf(d) : 0.0f;
    }
}

__global__ void norm_kernel(const long long* __restrict__ src,
                            const long long* __restrict__ dst,
                            const float* __restrict__ dis,
                            float* __restrict__ nrm, int e) {
    int i = blockIdx.x * blockDim.x + threadIdx.x;
    if (i < e) nrm[i] = -dis[(int)src[i]] * dis[(int)dst[i]];
}

__global__ void scale_copy_kernel(const float* __restrict__ in,
                                  float* __restrict__ out, float s, int n) {
    int i = blockIdx.x * blockDim.x + threadIdx.x;
    if (i < n) out[i] = s * in[i];
}

// ---------------- edge scatter: out[dst] += alpha*norm[e]*in[src] ----------------
// One wave32 per edge; each lane moves F/32 contiguous floats.

template <int F>
__global__ void prop_kernel(const long long* __restrict__ src,
                            const long long* __restrict__ dst,
                            const float* __restrict__ nrm,
                            const float* __restrict__ in,
                            float* __restrict__ out,
                            float alpha, int nEdges) {
    int gtid = blockIdx.x * blockDim.x + threadIdx.x;
    int edge = gtid >> 5;          // wave32: 32 lanes per edge
    int lane = threadIdx.x & 31;
    if (edge >= nEdges) return;

    int s = (int)src[edge];
    int d = (int)dst[edge];
    float w = alpha * nrm[edge];

    constexpr int VPL = F / 32;    // floats per lane (4 for F=128, 2 for F=64)
    const float* pin = in + (size_t)s * F + lane * VPL;
    float* pout      = out + (size_t)d * F + lane * VPL;

    if constexpr (VPL == 4) {
        float4 v = *reinterpret_cast<const float4*>(pin);
        atomicAdd(pout + 0, w * v.x);
        atomicAdd(pout + 1, w * v.y);
        atomicAdd(pout + 2, w * v.z);
        atomicAdd(pout + 3, w * v.w);
    } else {
        float2 v = *reinterpret_cast<const float2*>(pin);
        atomicAdd(pout + 0, w * v.x);
        atomicAdd(pout + 1, w * v.y);
    }
}

// ---------------- fused Chebyshev GEMM via fp32 WMMA ----------------
// out[m, :FOUT] = sum_{k=0..2} Tk[m,:] @ W[k,:,:] + bias, optional ReLU.
// Block: 128 threads = 4 waves; blockIdx.x picks a 16-row M tile (100000 = 6250*16),
// wave w owns output columns [16w, 16w+16). FOUT_PAD = FOUT rounded to 16.

template <int FIN, int FOUT, int FOUT_PAD, bool RELU>
__global__ void cheb_gemm_kernel(const float* __restrict__ T0,
                                 const float* __restrict__ T1,
                                 const float* __restrict__ T2,
                                 const float* __restrict__ W,    // [3][FIN][FOUT]
                                 const float* __restrict__ bias, // [FOUT]
                                 float* __restrict__ out) {      // [N][FOUT]
    const int waveInBlk = threadIdx.x >> 5;
    const int lane      = threadIdx.x & 31;
    const int half      = lane >> 4;   // 0: lanes 0-15, 1: lanes 16-31
    const int idx       = lane & 15;

    const int mbase = blockIdx.x * 16;
    const int nbase = waveInBlk * 16;
    if (nbase >= FOUT_PAD) return;     // wave-uniform: EXEC stays all-1s inside

    const int col   = nbase + idx;
    const bool colOK = (col < FOUT);

    v8f acc = {};
    const float* Ts[3] = {T0, T1, T2};

    #pragma unroll
    for (int k = 0; k < 3; ++k) {
        // A fragment source: row (mbase+idx) of Tk
        const float* A  = Ts[k] + (size_t)(mbase + idx) * FIN;
        const float* Bm = W + (size_t)k * FIN * FOUT;
        #pragma unroll 4
        for (int kk = 0; kk < FIN; kk += 4) {
            const int ka = kk + 2 * half;
            // A 16x4 f32 fragment: VGPR0 = K{0|2}, VGPR1 = K{1|3} (doc'd layout)
            v2f a;
            a.x = A[ka];
            a.y = A[ka + 1];
            // B 4x16 f32 fragment: lane = column, mirrored K-halves
            v2f b;
            b.x = colOK ? Bm[(size_t)ka * FOUT + col] : 0.0f;
            b.y = colOK ? Bm[(size_t)(ka + 1) * FOUT + col] : 0.0f;
            acc = __builtin_amdgcn_wmma_f32_16x16x4_f32(
                /*neg_a=*/false, a, /*neg_b=*/false, b,
                /*c_mod=*/(short)0, acc, /*reuse_a=*/false, /*reuse_b=*/false);
        }
    }

    if (colOK) {
        const float bb = bias[col];
        #pragma unroll
        for (int v = 0; v < 8; ++v) {
            // C/D layout: VGPR v, lane -> row (v + 8*half), col (lane&15)
            const int m = mbase + v + 8 * half;
            float val = acc[v] + bb;
            if (RELU) val = fmaxf(val, 0.0f);
            out[(size_t)m * FOUT + col] = val;
        }
    }
}

// ---------------- in-place row-wise log-softmax ----------------

__global__ void log_softmax_kernel(float* __restrict__ out, int n, int C) {
    int i = blockIdx.x * blockDim.x + threadIdx.x;
    if (i >= n) return;
    float* row = out + (size_t)i * C;
    float m = -INFINITY;
    for (int j = 0; j < C; ++j) m = fmaxf(m, row[j]);
    float s = 0.0f;
    for (int j = 0; j < C; ++j) s += expf(row[j] - m);
    float ls = logf(s);
    for (int j = 0; j < C; ++j) row[j] = row[j] - m - ls;
}

// ---------------- launch ----------------

extern "C" void kernel_launch(void* const* d_in, const int* in_sizes, int n_in,
                              void* d_out, int out_size, void* d_ws, size_t ws_size,
                              hipStream_t stream) {
    const float* x        = (const float*)d_in[0];        // [N,128]
    const long long* ei   = (const long long*)d_in[1];    // [2,E] int64
    const float* W1       = (const float*)d_in[2];        // [3,128,64]
    const float* b1       = (const float*)d_in[3];        // [64]
    const float* W2       = (const float*)d_in[4];        // [3,64,40]
    const float* b2       = (const float*)d_in[5];        // [40]
    float* out            = (float*)d_out;                // [N,40]

    const long long* src = ei;
    const long long* dst = ei + N_EDGES;

    // workspace carve-up (fp32)
    float* ws   = (float*)d_ws;
    float* deg  = ws;                               // N
    float* dis  = deg  + N_NODES;                   // N
    float* nrm  = dis  + N_NODES;                   // E
    float* tx1a = nrm  + N_EDGES;                   // N*128
    float* tx2a = tx1a + (size_t)N_NODES * 128;     // N*128
    float* h    = tx2a + (size_t)N_NODES * 128;     // N*64
    float* tx1b = h    + (size_t)N_NODES * 64;      // N*64
    float* tx2b = tx1b + (size_t)N_NODES * 64;      // N*64

    // zero the atomic accumulators
    hipMemsetAsync(deg,  0, (size_t)N_NODES * sizeof(float), stream);
    hipMemsetAsync(tx1a, 0, (size_t)N_NODES * 128 * sizeof(float), stream);
    hipMemsetAsync(tx1b, 0, (size_t)N_NODES * 64 * sizeof(float), stream);

    const int TB = 256;
    const int eBlocks = (N_EDGES + TB - 1) / TB;
    const int nBlocks = (N_NODES + TB - 1) / TB;
    const int propBlocks = (N_EDGES + (TB / 32) - 1) / (TB / 32); // 1 wave per edge

    // normalization
    deg_kernel <<<eBlocks, TB, 0, stream>>>(dst, deg, N_EDGES);
    dis_kernel <<<nBlocks, TB, 0, stream>>>(deg, dis, N_NODES);
    norm_kernel<<<eBlocks, TB, 0, stream>>>(src, dst, dis, nrm, N_EDGES);

    // ---- layer 1: 128 -> 64 ----
    prop_kernel<128><<<propBlocks, TB, 0, stream>>>(src, dst, nrm, x, tx1a, 1.0f, N_EDGES);
    {
        int n = N_NODES * 128;
        scale_copy_kernel<<<(n + TB - 1) / TB, TB, 0, stream>>>(x, tx2a, -1.0f, n);
    }
    prop_kernel<128><<<propBlocks, TB, 0, stream>>>(src, dst, nrm, tx1a, tx2a, 2.0f, N_EDGES);
    cheb_gemm_kernel<128, 64, 64, true>
        <<<N_NODES / 16, 128, 0, stream>>>(x, tx1a, tx2a, W1, b1, h);

    // ---- layer 2: 64 -> 40 ----
    prop_kernel<64><<<propBlocks, TB, 0, stream>>>(src, dst, nrm, h, tx1b, 1.0f, N_EDGES);
    {
        int n = N_NODES * 64;
        scale_copy_kernel<<<(n + TB - 1) / TB, TB, 0, stream>>>(h, tx2b, -1.0f, n);
    }
    prop_kernel<64><<<propBlocks, TB, 0, stream>>>(src, dst, nrm, tx1b, tx2b, 2.0f, N_EDGES);
    cheb_gemm_kernel<64, 40, 48, false>
        <<<N_NODES / 16, 128, 0, stream>>>(h, tx1b, tx2b, W2, b2, out);

    // ---- log-softmax over 40 classes, in place on d_out ----
    log_softmax_kernel<<<nBlocks, TB, 0, stream>>>(out, N_NODES, 40);
}